// ConstantRateTerm_4501125726472
// MI455X (gfx1250) — compile-verified
//
#include <hip/hip_runtime.h>
#include <cstdint>

// Problem geometry (reference: N_SPECIES=1024, N_T1=20000, N_T2=60000, BATCH=1024)
constexpr int NSP    = 1024;   // species per row (power of two, used in index math)
constexpr int BCOLS  = 8;      // batch columns per workgroup (lanes 0..7 of each stream)
constexpr int BLOCK  = 256;    // 8 wave32s -> 32 concurrent term streams per WG
constexpr int NSTRM  = BLOCK / BCOLS;  // 32 term streams

__global__ __launch_bounds__(BLOCK)
void ConstantRateTerm_kernel(const float* __restrict__ y,
                             const float* __restrict__ rates1,
                             const float* __restrict__ rates2,
                             const float* __restrict__ den_norm,
                             const int*   __restrict__ ir1,
                             const int*   __restrict__ ir2a,
                             const int*   __restrict__ ir2b,
                             const int*   __restrict__ io1,
                             const int*   __restrict__ io2,
                             float*       __restrict__ out,
                             int t1, int t2)
{
    // Transposed tiles: element (species n, batch col b) at [n*BCOLS + b].
    __shared__ float sY[NSP * BCOLS];   // 32 KB
    __shared__ float sO[NSP * BCOLS];   // 32 KB

    const int tid  = threadIdx.x;
    const int row0 = blockIdx.x * BCOLS;           // first batch row of this WG

    // ---- zero the output tile (plain ds stores) ----
    #pragma unroll
    for (int i = tid; i < NSP * BCOLS; i += BLOCK)
        sO[i] = 0.0f;

    // ---- async global -> LDS load of y tile, transposing on the fly ----
    // i enumerates (b, n); consecutive lanes take consecutive n -> coalesced reads.
    // Per-lane independent global AND LDS addresses let the async unit transpose.
    #pragma unroll
    for (int i = tid; i < NSP * BCOLS; i += BLOCK) {
        const int b = i >> 10;              // i / NSP
        const int n = i & (NSP - 1);        // i % NSP
        const unsigned long long gaddr =
            (unsigned long long)(y + (size_t)(row0 + b) * NSP + n);
        const unsigned ldsa = (unsigned)(size_t)&sY[n * BCOLS + b];
        asm volatile("global_load_async_to_lds_b32 %0, %1, off"
                     :: "v"(ldsa), "v"(gaddr) : "memory");
    }
    asm volatile("s_wait_asynccnt 0" ::: "memory");
    __syncthreads();

    // ---- term processing ----
    // stream id s = tid/8 in [0,32): which term this group of 8 lanes works on
    // lane's batch column b = tid%8 -> 8 consecutive, conflict-free LDS addresses
    const int s = tid >> 3;
    const int b = tid & (BCOLS - 1);
    const float dn = den_norm[0];

    // 1st-order terms: out[o] += k * y[a]
    #pragma unroll 4
    for (int t = s; t < t1; t += NSTRM) {
        const int   a = ir1[t];
        const int   o = io1[t];
        const float k = rates1[t];
        const float v = sY[a * BCOLS + b] * k;
        atomicAdd(&sO[o * BCOLS + b], v);           // ds_add_f32, conflict-free
    }

    // 2nd-order terms: out[o] += dn * k * y[a] * y[c]
    #pragma unroll 4
    for (int t = s; t < t2; t += NSTRM) {
        const int   a = ir2a[t];
        const int   c = ir2b[t];
        const int   o = io2[t];
        const float k = rates2[t] * dn;
        const float v = sY[a * BCOLS + b] * sY[c * BCOLS + b] * k;
        atomicAdd(&sO[o * BCOLS + b], v);           // ds_add_f32, conflict-free
    }

    __syncthreads();

    // ---- async LDS -> global store of the finished tile (transpose back) ----
    #pragma unroll
    for (int i = tid; i < NSP * BCOLS; i += BLOCK) {
        const int bb = i >> 10;
        const int n  = i & (NSP - 1);
        const unsigned long long gaddr =
            (unsigned long long)(out + (size_t)(row0 + bb) * NSP + n);
        const unsigned ldsa = (unsigned)(size_t)&sO[n * BCOLS + bb];
        asm volatile("global_store_async_from_lds_b32 %0, %1, off"
                     :: "v"(gaddr), "v"(ldsa) : "memory");
    }
    asm volatile("s_wait_asynccnt 0" ::: "memory");
}

extern "C" void kernel_launch(void* const* d_in, const int* in_sizes, int n_in,
                              void* d_out, int out_size, void* d_ws, size_t ws_size,
                              hipStream_t stream)
{
    // setup_inputs() order:
    // 0:t_in 1:y_in 2:rates_1st 3:rates_2nd 4:den_norm
    // 5:inds_r1 6:inds_r2a 7:inds_r2b 8:inds_out1 9:inds_out2
    const float* y    = (const float*)d_in[1];
    const float* r1   = (const float*)d_in[2];
    const float* r2   = (const float*)d_in[3];
    const float* dn   = (const float*)d_in[4];
    const int*   ir1  = (const int*)d_in[5];
    const int*   ir2a = (const int*)d_in[6];
    const int*   ir2b = (const int*)d_in[7];
    const int*   io1  = (const int*)d_in[8];
    const int*   io2  = (const int*)d_in[9];
    float*       out  = (float*)d_out;

    const int t1    = in_sizes[2];
    const int t2    = in_sizes[3];
    const int batch = in_sizes[1] / NSP;

    dim3 grid(batch / BCOLS);   // 128 workgroups, each owns 8 full output rows
    dim3 block(BLOCK);
    ConstantRateTerm_kernel<<<grid, block, 0, stream>>>(
        y, r1, r2, dn, ir1, ir2a, ir2b, io1, io2, out, t1, t2);
}